// Sinkhorn_75600014344218
// MI455X (gfx1250) — compile-verified
//
#include <hip/hip_runtime.h>
#include <stdint.h>
#include <math.h>

namespace {

constexpr int B = 64;
constexpr int M = 512;
constexpr int N = 512;
constexpr int ITERS = 20;
constexpr float NEG_INV_L = -500.0f;   // -1/lambda, lambda = 0.002

typedef float v2f __attribute__((ext_vector_type(2)));
typedef float v8f __attribute__((ext_vector_type(8)));

// merge two online-logsumexp states (mx,s) <- (mx,s) ⊕ (mo,so)
__device__ __forceinline__ void lse_merge(float& mx, float& s, float mo, float so) {
  float mn = fmaxf(mx, mo);
  s = s * __expf(mx - mn) + so * __expf(mo - mn);
  mx = mn;
}

__global__ void init_kernel(const float* __restrict__ a, const float* __restrict__ b,
                            float* __restrict__ U, float* __restrict__ logA,
                            float* __restrict__ logB) {
  int i = blockIdx.x * blockDim.x + threadIdx.x;  // over B*M (== B*N)
  U[i]    = 0.0f;
  logA[i] = logf(a[i]);
  logB[i] = logf(b[i]);
}

// V[b,n] = logsumexp_m( c[b,m,n]*(-500) - U[b,m] ) - logB[b,n]
// block (32 lanes, 16 waves); wave owns 32 cols x 32 rows.
// Pipeline: 8 rows/step staged by two async b128 copies (512 B each), double buffered.
// Final 16-partial combine: max-normalize, then ones-matrix WMMA contraction.
__global__ __launch_bounds__(512) void col_kernel(const float* __restrict__ c,
                                                  const float* __restrict__ U,
                                                  const float* __restrict__ logB,
                                                  float* __restrict__ V) {
  // [buffer][wave][row 0..7][col 0..31]
  __shared__ __align__(16) float buf[2][16][8][32];   // 32 KB
  __shared__ float cm[16][32];
  __shared__ float cs[16][32];
  __shared__ float eb[16][32];                        // normalized partials

  const int lane = threadIdx.x;
  const int wv   = threadIdx.y;
  const int b    = blockIdx.y;
  const int n0   = blockIdx.x * 32;
  const int m0   = wv * 32;

  // this wave's 32 U values; broadcast per-row via __shfl (wave32)
  float uvals = U[(size_t)b * M + m0 + lane];

  // async source: lane covers row (lane>>3), 16-byte chunk (lane&7) -> 4 rows / 512 B per op
  const uint64_t rowstride = (uint64_t)N * sizeof(float);
  uint64_t ga = (uint64_t)(uintptr_t)(c + ((size_t)b * M + m0) * N + n0)
              + (uint64_t)(lane >> 3) * rowstride + (uint64_t)(lane & 7) * 16u;

  // LDS destinations (flat addr low 32 bits == LDS offset); halves 512 B apart
  unsigned ldsA[2], ldsB[2];
  ldsA[0] = (unsigned)(uintptr_t)(&buf[0][wv][0][0] + lane * 4);
  ldsB[0] = (unsigned)(uintptr_t)(&buf[0][wv][4][0] + lane * 4);
  ldsA[1] = (unsigned)(uintptr_t)(&buf[1][wv][0][0] + lane * 4);
  ldsB[1] = (unsigned)(uintptr_t)(&buf[1][wv][4][0] + lane * 4);

  // prologue: stage step 0 (rows 0..7) into buffer 0
  asm volatile("global_load_async_to_lds_b128 %0, %1, off"
               :: "v"(ldsA[0]), "v"(ga) : "memory");
  asm volatile("global_load_async_to_lds_b128 %0, %1, off"
               :: "v"(ldsB[0]), "v"(ga + 4 * rowstride) : "memory");

  float mx = -INFINITY, s = 0.0f;
  #pragma unroll
  for (int i = 0; i < 4; ++i) {          // 4 steps x 8 rows = 32 rows
    if (i < 3) {
      uint64_t gn = ga + (uint64_t)(i + 1) * 8 * rowstride;
      int k = (i + 1) & 1;
      asm volatile("global_load_async_to_lds_b128 %0, %1, off"
                   :: "v"(ldsA[k]), "v"(gn) : "memory");
      asm volatile("global_load_async_to_lds_b128 %0, %1, off"
                   :: "v"(ldsB[k]), "v"(gn + 4 * rowstride) : "memory");
      asm volatile("s_wait_asynccnt 0x2" ::: "memory");   // step i's two copies done
    } else {
      asm volatile("s_wait_asynccnt 0x0" ::: "memory");
    }

    // 8-element group: max first, then one exp per element + one rescale exp
    float x[8];
    #pragma unroll
    for (int r = 0; r < 8; ++r)
      x[r] = buf[i & 1][wv][r][lane] * NEG_INV_L - __shfl(uvals, i * 8 + r);

    float g = x[0];
    #pragma unroll
    for (int r = 1; r < 8; ++r) g = fmaxf(g, x[r]);

    float mn = fmaxf(mx, g);
    float e = 0.0f;
    #pragma unroll
    for (int r = 0; r < 8; ++r) e += __expf(x[r] - mn);
    s = s * __expf(mx - mn) + e;
    mx = mn;
  }

  cm[wv][lane] = mx;
  cs[wv][lane] = s;
  __syncthreads();

  if (wv == 0) {
    // column-wise max over the 16 wave partials (lane <-> column)
    float Mx = cm[0][lane];
    #pragma unroll
    for (int w = 1; w < 16; ++w) Mx = fmaxf(Mx, cm[w][lane]);

    // normalized partials E[w][n] = cs[w][n] * exp(cm[w][n] - Mx[n])
    #pragma unroll
    for (int w = 0; w < 16; ++w)
      eb[w][lane] = cs[w][lane] * __expf(cm[w][lane] - Mx);

    // S[n] = sum_w E[w][n] via ones-matrix WMMA: D[i,j] = sum_k B[k,j]
    // B column j lives in lanes {j, j+16}; with A == ones the K-slot permutation
    // is irrelevant (full sum over K). Two chains: columns 0..15 and 16..31.
    const int col  = lane & 15;
    const int half = lane >> 4;
    v2f ones; ones.x = 1.0f; ones.y = 1.0f;
    v8f acc0 = {}; v8f acc1 = {};
    #pragma unroll
    for (int t = 0; t < 4; ++t) {        // 4 K-chunks x 4 partials = 16
      v2f b0, b1;
      b0.x = eb[4 * t + half][col];      b0.y = eb[4 * t + 2 + half][col];
      b1.x = eb[4 * t + half][16 + col]; b1.y = eb[4 * t + 2 + half][16 + col];
      acc0 = __builtin_amdgcn_wmma_f32_16x16x4_f32(false, ones, false, b0,
                                                   (short)0, acc0, false, false);
      acc1 = __builtin_amdgcn_wmma_f32_16x16x4_f32(false, ones, false, b1,
                                                   (short)0, acc1, false, false);
    }
    // D rows are identical; VGPR0 of D holds column (lane&15) in every lane.
    float S = (lane < 16) ? acc0[0] : acc1[0];
    V[(size_t)b * N + n0 + lane] = Mx + logf(S) - logB[(size_t)b * N + n0 + lane];
  }
}

// U[b,m] = logsumexp_n( c[b,m,n]*(-500) - V[b,n] ) - logA[b,m]
// one wave per row; lane holds its 16 values in registers -> exact max, 1 exp/element
__global__ __launch_bounds__(512) void row_kernel(const float* __restrict__ c,
                                                  const float* __restrict__ V,
                                                  const float* __restrict__ logA,
                                                  float* __restrict__ U) {
  const int lane = threadIdx.x;
  const int b    = blockIdx.y;
  const int m    = blockIdx.x * 16 + threadIdx.y;

  const float4* crow = (const float4*)(c + ((size_t)b * M + m) * N);
  const float4* vrow = (const float4*)(V + (size_t)b * N);

  float x[16];
  #pragma unroll
  for (int j = 0; j < 4; ++j) {          // 128 float4s per row / 32 lanes
    float4 cv = crow[j * 32 + lane];
    float4 vv = vrow[j * 32 + lane];
    x[j * 4 + 0] = cv.x * NEG_INV_L - vv.x;
    x[j * 4 + 1] = cv.y * NEG_INV_L - vv.y;
    x[j * 4 + 2] = cv.z * NEG_INV_L - vv.z;
    x[j * 4 + 3] = cv.w * NEG_INV_L - vv.w;
  }

  float mx = x[0];
  #pragma unroll
  for (int k = 1; k < 16; ++k) mx = fmaxf(mx, x[k]);
  float s = 0.0f;
  #pragma unroll
  for (int k = 0; k < 16; ++k) s += __expf(x[k] - mx);

  #pragma unroll
  for (int off = 16; off > 0; off >>= 1) {   // wave32 butterfly LSE-merge
    float mo = __shfl_xor(mx, off);
    float so = __shfl_xor(s, off);
    lse_merge(mx, s, mo, so);
  }

  if (lane == 0)
    U[(size_t)b * M + m] = mx + logf(s) - logA[(size_t)b * M + m];
}

// out = exp( c*(-500) - U[b,m] - V[b,n] )
__global__ void final_kernel(const float* __restrict__ c, const float* __restrict__ U,
                             const float* __restrict__ V, float* __restrict__ out) {
  size_t q = (size_t)blockIdx.x * blockDim.x + threadIdx.x;  // over B*M*N/4
  size_t e = q * 4;
  int n = (int)(e % N);
  int m = (int)((e / N) % M);
  int b = (int)(e / ((size_t)N * M));

  float4 cv = ((const float4*)c)[q];
  float  u  = U[(size_t)b * M + m];
  float4 vv = *(const float4*)(V + (size_t)b * N + n);

  float4 o;
  o.x = __expf(cv.x * NEG_INV_L - u - vv.x);
  o.y = __expf(cv.y * NEG_INV_L - u - vv.y);
  o.z = __expf(cv.z * NEG_INV_L - u - vv.z);
  o.w = __expf(cv.w * NEG_INV_L - u - vv.w);
  ((float4*)out)[q] = o;
}

}  // namespace

extern "C" void kernel_launch(void* const* d_in, const int* in_sizes, int n_in,
                              void* d_out, int out_size, void* d_ws, size_t ws_size,
                              hipStream_t stream) {
  const float* c = (const float*)d_in[0];
  const float* a = (const float*)d_in[1];
  const float* b = (const float*)d_in[2];
  float* out = (float*)d_out;

  // workspace: U[B*M], V[B*N], logA[B*M], logB[B*N]  (512 KB total)
  float* U    = (float*)d_ws;
  float* V    = U + (size_t)B * M;
  float* logA = V + (size_t)B * N;
  float* logB = logA + (size_t)B * M;

  init_kernel<<<(B * M) / 256, 256, 0, stream>>>(a, b, U, logA, logB);

  for (int t = 0; t < ITERS; ++t) {
    col_kernel<<<dim3(N / 32, B), dim3(32, 16), 0, stream>>>(c, U, logB, V);
    row_kernel<<<dim3(M / 16, B), dim3(32, 16), 0, stream>>>(c, V, logA, U);
  }

  final_kernel<<<(unsigned)((size_t)B * M * N / 4 / 256), 256, 0, stream>>>(c, U, V, out);
}